// CoupledFourierSystem_77567109365841
// MI455X (gfx1250) — compile-verified
//
#include <hip/hip_runtime.h>

typedef __attribute__((ext_vector_type(2))) float v2f;
typedef __attribute__((ext_vector_type(8))) float v8f;

#define S_TOTAL 32768
#define DIM     64
#define NTERM   32      // H * 2
#define ROWS    64      // time rows per block
#define TPITCH  33      // padded pitch for per-dim term arrays (gcd(33,64)=1 -> conflict free)
#define WPITCH  68      // padded pitch for W / pre_sol tiles (bank = 4*row+k, conflict free)

__global__ __launch_bounds__(256)
void CoupledFourierSystem_77567109365841_kernel(const float* __restrict__ s,
                                                const float* __restrict__ A,
                                                const float* __restrict__ phi,
                                                const float* __restrict__ w,
                                                const float* __restrict__ Wm,
                                                const float* __restrict__ bias,
                                                float* __restrict__ out)
{
    __shared__ float sS[ROWS];
    __shared__ float sA[DIM * TPITCH];
    __shared__ float sPhi[DIM * TPITCH];
    __shared__ float sW[DIM * TPITCH];
    __shared__ float sWmat[DIM * WPITCH];
    __shared__ float sBias[DIM];
    __shared__ float sPS[ROWS * WPITCH];

    const int tid  = threadIdx.x;
    const int row0 = blockIdx.x * ROWS;

    // ---------------- Stage inputs into LDS ----------------
    if (tid < ROWS) sS[tid] = s[row0 + tid];
    if (tid < DIM)  sBias[tid] = bias[tid];
    for (int i = tid; i < DIM * NTERM; i += 256) {
        const int d = i >> 5, t = i & 31;            // A/phi/w are [d][h][c] = d*32 + t
        sA[d * TPITCH + t]   = A[i];
        sPhi[d * TPITCH + t] = phi[i];
        sW[d * TPITCH + t]   = w[i];
    }
    for (int i = tid; i < DIM * DIM; i += 256) {
        const int j = i >> 6, d = i & 63;            // W[j][d]
        sWmat[j * WPITCH + d] = Wm[i];
    }
    __syncthreads();

    // ---- Phase A: pre_sol[r][d] = sum_t A[d,t] * cos(w[d,t]*s[r] + phi[d,t]) ----
    {
        const int d    = tid & (DIM - 1);
        const int rblk = tid >> 6;                   // 0..3, 16 rows each
        float sv[16];
        #pragma unroll
        for (int r = 0; r < 16; ++r) sv[r] = sS[rblk * 16 + r];
        float acc[16];
        #pragma unroll
        for (int r = 0; r < 16; ++r) acc[r] = 0.0f;

        #pragma unroll 4
        for (int t = 0; t < NTERM; ++t) {
            const float av = sA[d * TPITCH + t];
            const float pv = sPhi[d * TPITCH + t];
            const float wv = sW[d * TPITCH + t];
            #pragma unroll
            for (int r = 0; r < 16; ++r)
                acc[r] = fmaf(av, __cosf(fmaf(wv, sv[r], pv)), acc[r]);
        }
        #pragma unroll
        for (int r = 0; r < 16; ++r)
            sPS[(rblk * 16 + r) * WPITCH + d] = acc[r];
    }
    __syncthreads();

    // ---- Phase B: out[64x64] = pre_sol @ W^T + b  via V_WMMA_F32_16X16X4_F32 ----
    const int wave = tid >> 5;
    const int lane = tid & 31;
    const int l    = lane & 15;      // N (B/D) or M (A) position
    const int hi   = lane >> 4;      // upper half-lanes: K += 2 (A/B), M += 8 (D)

    for (int tile = wave; tile < 16; tile += 8) {    // uniform per wave: EXEC all ones
        const int mt = tile >> 2;                    // row-tile
        const int nt = tile & 3;                     // col-tile
        v8f c = {0.f, 0.f, 0.f, 0.f, 0.f, 0.f, 0.f, 0.f};
        #pragma unroll
        for (int kk = 0; kk < 16; ++kk) {
            const int kb = kk * 4 + 2 * hi;
            // A fragment: 16x4 f32, lane l holds row M=l, K={kb, kb+1}
            v2f a = *(const v2f*)&sPS[(mt * 16 + l) * WPITCH + kb];
            // B fragment: 4x16 f32 = W^T, lane l holds col N=l, K={kb, kb+1}
            //   W^T[k][n] = W[n][k]  ->  sWmat[n * WPITCH + k]
            v2f bfrag = *(const v2f*)&sWmat[(nt * 16 + l) * WPITCH + kb];
            c = __builtin_amdgcn_wmma_f32_16x16x4_f32(
                    false, a, false, bfrag, (short)0, c, false, false);
        }
        const float bv = sBias[nt * 16 + l];
        #pragma unroll
        for (int v = 0; v < 8; ++v) {
            // D layout: VGPR v -> M = v + 8*hi, N = l
            const int rg = row0 + mt * 16 + v + 8 * hi;
            out[rg * DIM + nt * 16 + l] = c[v] + bv;
        }
    }
}

extern "C" void kernel_launch(void* const* d_in, const int* in_sizes, int n_in,
                              void* d_out, int out_size, void* d_ws, size_t ws_size,
                              hipStream_t stream) {
    // setup_inputs order: s, x(unused), A, phi, w, W, b
    const float* s   = (const float*)d_in[0];
    const float* A   = (const float*)d_in[2];
    const float* phi = (const float*)d_in[3];
    const float* w   = (const float*)d_in[4];
    const float* Wm  = (const float*)d_in[5];
    const float* b   = (const float*)d_in[6];
    float* out = (float*)d_out;

    (void)in_sizes; (void)n_in; (void)out_size; (void)d_ws; (void)ws_size;

    const int blocks = S_TOTAL / ROWS;   // 512
    CoupledFourierSystem_77567109365841_kernel<<<blocks, 256, 0, stream>>>(
        s, A, phi, w, Wm, b, out);
}